// sa_layer_70970039599892
// MI455X (gfx1250) — compile-verified
//
#include <hip/hip_runtime.h>
#include <hip/hip_bf16.h>
#include <math.h>

// ---------------------------------------------------------------------------
// Geometry: x viewed as (8192, 4096) fp32; row r -> bg = r>>6, c = r&63.
// c < 32  : x0 branch (channel-gate),  c >= 32 : x1 branch (GN + pixel gate).
// ---------------------------------------------------------------------------
#define NROWS   8192
#define HW      4096
#define CS      32
#define NBG     128          // 32 batches * 4 groups
#define EPSF    1e-5f

typedef __attribute__((ext_vector_type(2))) float v2f;
typedef __attribute__((ext_vector_type(8))) float v8f;

__device__ __forceinline__ float sigmoidf_(float v) {
    return 1.0f / (1.0f + expf(-v));
}

// ---------------------------------------------------------------------------
// Kernel 1: per-row sum and sum-of-squares over the 4096 spatial elements.
// One block per row, 256 threads, float4 loads (16 elements / thread).
// ---------------------------------------------------------------------------
__global__ __launch_bounds__(256) void k_row_stats(const float* __restrict__ x,
                                                   float* __restrict__ mean,
                                                   float* __restrict__ sumsq) {
    const int r = blockIdx.x;
    const int t = threadIdx.x;
    const float4* xr = (const float4*)(x + (size_t)r * HW);
    float s = 0.0f, s2 = 0.0f;
#pragma unroll
    for (int i = 0; i < 4; ++i) {
        float4 v = xr[t + i * 256];
        s  += v.x + v.y + v.z + v.w;
        s2 += v.x * v.x + v.y * v.y + v.z * v.z + v.w * v.w;
    }
    __shared__ float ss[256];
    __shared__ float sq[256];
    ss[t] = s; sq[t] = s2;
    __syncthreads();
    for (int o = 128; o > 0; o >>= 1) {
        if (t < o) { ss[t] += ss[t + o]; sq[t] += sq[t + o]; }
        __syncthreads();
    }
    if (t == 0) {
        mean[r]  = ss[0] * (1.0f / (float)HW);
        sumsq[r] = sq[0];
    }
}

// ---------------------------------------------------------------------------
// Kernel 2: tiny per-row prep.
//  c<32 : sig0 = sigmoid(cw*mean + cb) ; p0 = mean*(1-sig0)   (exact identity:
//         x_reid_0 = x0*(1-sig0) so its spatial mean needs no extra pass)
//  c>=32: mu = mean ; rs = rsqrt(E[x^2]-mu^2 + eps)
// ---------------------------------------------------------------------------
__global__ __launch_bounds__(256) void k_prep(const float* __restrict__ mean,
                                              const float* __restrict__ sumsq,
                                              const float* __restrict__ cweight,
                                              const float* __restrict__ cbias,
                                              float* __restrict__ sig0,
                                              float* __restrict__ p0,
                                              float* __restrict__ muA,
                                              float* __restrict__ rsA) {
    const int r = blockIdx.x * blockDim.x + threadIdx.x;
    if (r >= NROWS) return;
    const int bg = r >> 6;
    const int c  = r & 63;
    const float m = mean[r];
    if (c < CS) {
        const int idx = bg * CS + c;
        const float s = sigmoidf_(cweight[c] * m + cbias[c]);
        sig0[idx] = s;
        p0[idx]   = m * (1.0f - s);
    } else {
        const int c1  = c - CS;
        const int idx = bg * CS + c1;
        const float var = sumsq[r] * (1.0f / (float)HW) - m * m;
        muA[idx] = m;
        rsA[idx] = rsqrtf(var + EPSF);
    }
}

// ---------------------------------------------------------------------------
// Kernel 3: p1[bg,c1] = mean_HW( x1 * (1 - sigmoid(sw*((x1-mu)*rs*gw+gb)+sb)) )
// One block per (bg,c1) pair of the x1 half.
// ---------------------------------------------------------------------------
__global__ __launch_bounds__(256) void k_p1_reduce(const float* __restrict__ x,
                                                   const float* __restrict__ sweight,
                                                   const float* __restrict__ sbias,
                                                   const float* __restrict__ gn_w,
                                                   const float* __restrict__ gn_b,
                                                   const float* __restrict__ muA,
                                                   const float* __restrict__ rsA,
                                                   float* __restrict__ p1) {
    const int idx = blockIdx.x;           // bg*32 + c1
    const int bg  = idx >> 5;
    const int c1  = idx & 31;
    const int r   = bg * 64 + CS + c1;
    const int t   = threadIdx.x;
    const float mu = muA[idx], rs = rsA[idx];
    const float gw = gn_w[c1], gb = gn_b[c1], sb = sbias[c1];
    const float4* xr  = (const float4*)(x + (size_t)r * HW);
    const float4* swr = (const float4*)(sweight + (size_t)c1 * HW);
    float acc = 0.0f;
#pragma unroll
    for (int i = 0; i < 4; ++i) {
        float4 xv = xr[t + i * 256];
        float4 sw = swr[t + i * 256];
        float tx, sg;
        tx = sw.x * ((xv.x - mu) * rs * gw + gb) + sb; sg = sigmoidf_(tx); acc += xv.x * (1.0f - sg);
        tx = sw.y * ((xv.y - mu) * rs * gw + gb) + sb; sg = sigmoidf_(tx); acc += xv.y * (1.0f - sg);
        tx = sw.z * ((xv.z - mu) * rs * gw + gb) + sb; sg = sigmoidf_(tx); acc += xv.z * (1.0f - sg);
        tx = sw.w * ((xv.w - mu) * rs * gw + gb) + sb; sg = sigmoidf_(tx); acc += xv.w * (1.0f - sg);
    }
    __shared__ float ss[256];
    ss[t] = acc;
    __syncthreads();
    for (int o = 128; o > 0; o >>= 1) {
        if (t < o) ss[t] += ss[t + o];
        __syncthreads();
    }
    if (t == 0) p1[idx] = ss[0] * (1.0f / (float)HW);
}

// ---------------------------------------------------------------------------
// Kernel 4: distill MLP on a (128 x 32) matrix via V_WMMA_F32_16X16X4_F32.
//   h = p @ fc1_w^T + fc1_b ; LayerNorm(32) ; ReLU ; o = h @ fc2_w^T + fc2_b
//   g = sigmoid(o)
// 512 threads = 16 waves; wave w owns output tile (tm = w/2, tn = w%2),
// K=32 consumed as 8 steps of K=4.  Operand layout per ISA 7.12.2:
//   lane L : A[L%16][(L/16)*2 + v],  B[(L/16)*2 + v][L%16],
//   D lane L vgpr v : row tm*16 + (L/16)*8 + v, col tn*16 + L%16.
// blockIdx.x selects (p0 -> g0) or (p1 -> g1).
// ---------------------------------------------------------------------------
#define LNSTRIDE 34   // LDS row stride in floats: keeps 8B alignment, no 16-lane bank conflict

__global__ __launch_bounds__(512) void k_distill(const float* __restrict__ p0,
                                                 const float* __restrict__ p1,
                                                 const float* __restrict__ fc1_w,
                                                 const float* __restrict__ fc1_b,
                                                 const float* __restrict__ ln_w,
                                                 const float* __restrict__ ln_b,
                                                 const float* __restrict__ fc2_w,
                                                 const float* __restrict__ fc2_b,
                                                 float* __restrict__ g0,
                                                 float* __restrict__ g1) {
    const float* p = (blockIdx.x == 0) ? p0 : p1;
    float*       g = (blockIdx.x == 0) ? g0 : g1;

    __shared__ float sh[NBG * LNSTRIDE];

    const int t  = threadIdx.x;
    const int w  = t >> 5;        // wave id 0..15
    const int L  = t & 31;        // lane id
    const int tm = w >> 1;        // M tile 0..7
    const int tn = w & 1;         // N tile 0..1
    const int Lm = L & 15;
    const int kb = (L >> 4) * 2;  // k base within K=4 step
    const int Arow = tm * 16 + Lm;    // A-matrix row for this lane
    const int Brow = tn * 16 + Lm;    // weight row (output feature) for B operand
    const int N    = tn * 16 + Lm;    // D column for this lane

    // ---- fc1: D = p(128x32) x fc1_w^T(32x32) ----
    v8f acc = {};
#pragma unroll
    for (int kc = 0; kc < 8; ++kc) {
        const int k = kc * 4 + kb;
        v2f a, b;
        a.x = p[Arow * CS + k];
        a.y = p[Arow * CS + k + 1];
        b.x = fc1_w[Brow * CS + k];      // B[k][n] = fc1_w[n][k]
        b.y = fc1_w[Brow * CS + k + 1];
        acc = __builtin_amdgcn_wmma_f32_16x16x4_f32(false, a, false, b,
                                                    (short)0, acc, false, false);
    }
#pragma unroll
    for (int v = 0; v < 8; ++v) {
        const int M = tm * 16 + (L >> 4) * 8 + v;
        sh[M * LNSTRIDE + N] = acc[v] + fc1_b[N];
    }
    __syncthreads();

    // ---- LayerNorm over 32 features + ReLU (one row per thread 0..127) ----
    if (t < NBG) {
        float s = 0.0f, s2 = 0.0f;
#pragma unroll
        for (int j = 0; j < CS; ++j) { float v = sh[t * LNSTRIDE + j]; s += v; s2 += v * v; }
        const float mu  = s * (1.0f / (float)CS);
        const float var = s2 * (1.0f / (float)CS) - mu * mu;
        const float rs  = rsqrtf(var + EPSF);
#pragma unroll
        for (int j = 0; j < CS; ++j) {
            float v = (sh[t * LNSTRIDE + j] - mu) * rs * ln_w[j] + ln_b[j];
            sh[t * LNSTRIDE + j] = v > 0.0f ? v : 0.0f;
        }
    }
    __syncthreads();

    // ---- fc2 + sigmoid ----
    acc = {};
#pragma unroll
    for (int kc = 0; kc < 8; ++kc) {
        const int k = kc * 4 + kb;
        v2f a, b;
        a.x = sh[Arow * LNSTRIDE + k];
        a.y = sh[Arow * LNSTRIDE + k + 1];
        b.x = fc2_w[Brow * CS + k];
        b.y = fc2_w[Brow * CS + k + 1];
        acc = __builtin_amdgcn_wmma_f32_16x16x4_f32(false, a, false, b,
                                                    (short)0, acc, false, false);
    }
#pragma unroll
    for (int v = 0; v < 8; ++v) {
        const int M = tm * 16 + (L >> 4) * 8 + v;
        g[M * CS + N] = sigmoidf_(acc[v] + fc2_b[N]);
    }
}

// ---------------------------------------------------------------------------
// Kernel 5: fused final elementwise pass.
//  c<32 : out = x0 * (sig0 + (1-sig0)*g0)                  (pure scale per row)
//  c>=32: s = sigmoid(sw*((x1-mu)*rs*gw+gb)+sb)
//         out = x1 * (s + (1-s)*g1)
// ---------------------------------------------------------------------------
__global__ __launch_bounds__(256) void k_final(const float* __restrict__ x,
                                               const float* __restrict__ sweight,
                                               const float* __restrict__ sbias,
                                               const float* __restrict__ gn_w,
                                               const float* __restrict__ gn_b,
                                               const float* __restrict__ sig0,
                                               const float* __restrict__ muA,
                                               const float* __restrict__ rsA,
                                               const float* __restrict__ g0,
                                               const float* __restrict__ g1,
                                               float* __restrict__ out) {
    const int r  = blockIdx.x;
    const int bg = r >> 6;
    const int c  = r & 63;
    const int t  = threadIdx.x;
    const float4* xr = (const float4*)(x + (size_t)r * HW);
    float4* outr     = (float4*)(out + (size_t)r * HW);

    if (c < CS) {
        const int idx = bg * CS + c;
        const float sg    = sig0[idx];
        const float scale = sg + (1.0f - sg) * g0[idx];
#pragma unroll
        for (int i = 0; i < 4; ++i) {
            float4 v = xr[t + i * 256];
            v.x *= scale; v.y *= scale; v.z *= scale; v.w *= scale;
            outr[t + i * 256] = v;
        }
    } else {
        const int c1  = c - CS;
        const int idx = bg * CS + c1;
        const float mu = muA[idx], rs = rsA[idx];
        const float gw = gn_w[c1], gb = gn_b[c1], sb = sbias[c1];
        const float g1v = g1[idx];
        const float4* swr = (const float4*)(sweight + (size_t)c1 * HW);
#pragma unroll
        for (int i = 0; i < 4; ++i) {
            float4 xv = xr[t + i * 256];
            float4 sw = swr[t + i * 256];
            float tx, s;
            tx = sw.x * ((xv.x - mu) * rs * gw + gb) + sb; s = sigmoidf_(tx); xv.x *= (s + (1.0f - s) * g1v);
            tx = sw.y * ((xv.y - mu) * rs * gw + gb) + sb; s = sigmoidf_(tx); xv.y *= (s + (1.0f - s) * g1v);
            tx = sw.z * ((xv.z - mu) * rs * gw + gb) + sb; s = sigmoidf_(tx); xv.z *= (s + (1.0f - s) * g1v);
            tx = sw.w * ((xv.w - mu) * rs * gw + gb) + sb; s = sigmoidf_(tx); xv.w *= (s + (1.0f - s) * g1v);
            outr[t + i * 256] = xv;
        }
    }
}

// ---------------------------------------------------------------------------
// Launch
// ---------------------------------------------------------------------------
extern "C" void kernel_launch(void* const* d_in, const int* in_sizes, int n_in,
                              void* d_out, int out_size, void* d_ws, size_t ws_size,
                              hipStream_t stream) {
    const float* x       = (const float*)d_in[0];
    const float* cweight = (const float*)d_in[1];
    const float* cbias   = (const float*)d_in[2];
    const float* sweight = (const float*)d_in[3];
    const float* sbias   = (const float*)d_in[4];
    const float* gn_w    = (const float*)d_in[5];
    const float* gn_b    = (const float*)d_in[6];
    const float* fc1_w   = (const float*)d_in[7];
    const float* fc1_b   = (const float*)d_in[8];
    const float* ln_w    = (const float*)d_in[9];
    const float* ln_b    = (const float*)d_in[10];
    const float* fc2_w   = (const float*)d_in[11];
    const float* fc2_b   = (const float*)d_in[12];
    float* out = (float*)d_out;

    // workspace layout (floats)
    float* ws    = (float*)d_ws;
    float* mean  = ws;            // 8192
    float* sumsq = ws + 8192;     // 8192
    float* sig0  = ws + 16384;    // 4096
    float* muA   = ws + 20480;    // 4096
    float* rsA   = ws + 24576;    // 4096
    float* p0    = ws + 28672;    // 4096
    float* p1    = ws + 32768;    // 4096
    float* g0    = ws + 36864;    // 4096
    float* g1    = ws + 40960;    // 4096

    k_row_stats<<<NROWS, 256, 0, stream>>>(x, mean, sumsq);
    k_prep<<<NROWS / 256, 256, 0, stream>>>(mean, sumsq, cweight, cbias,
                                            sig0, p0, muA, rsA);
    k_p1_reduce<<<NBG * CS, 256, 0, stream>>>(x, sweight, sbias, gn_w, gn_b,
                                              muA, rsA, p1);
    k_distill<<<2, 512, 0, stream>>>(p0, p1, fc1_w, fc1_b, ln_w, ln_b,
                                     fc2_w, fc2_b, g0, g1);
    k_final<<<NROWS, 256, 0, stream>>>(x, sweight, sbias, gn_w, gn_b,
                                       sig0, muA, rsA, g0, g1, out);
}